// LightGCN_67688684585004
// MI455X (gfx1250) — compile-verified
//
#include <hip/hip_runtime.h>
#include <hip/hip_bf16.h>
#include <stdint.h>

#define GCN_N_USERS 100000
#define GCN_N_ITEMS 400000
#define GCN_N_NODES (GCN_N_USERS + GCN_N_ITEMS)
#define GCN_EMB 64
#define GCN_CHUNK 256           // edges staged per block iteration
#define GCN_SCATTER_BLOCK 256   // 8 waves (wave32)

// ---------------- CDNA5 async global->LDS path (guarded) ----------------
#if __has_builtin(__builtin_amdgcn_global_load_async_to_lds_b32) && \
    __has_builtin(__builtin_amdgcn_s_wait_asynccnt)
#define GCN_ASYNC_LDS 1
#else
#define GCN_ASYNC_LDS 0
#endif

#if GCN_ASYNC_LDS
// Signature per hipcc diagnostic: (addrspace(1) int*, addrspace(3) int*, imm, imm)
typedef __attribute__((address_space(1))) int glb_i32_t;
typedef __attribute__((address_space(3))) int lds_i32_t;
__device__ __forceinline__ void gcn_async_b32(const void* gsrc, void* ldst) {
  __builtin_amdgcn_global_load_async_to_lds_b32(
      (glb_i32_t*)gsrc, (lds_i32_t*)ldst, 0, 0);
}
#endif

// Split workgroup barrier WITHOUT the acq-rel fence __syncthreads() implies.
// Avoids the per-chunk `s_wait_storecnt 0` drain of in-flight L2 atomics seen
// in the previous build's loop latch. LDS producer/consumer ordering is
// guaranteed separately: producers wait ASYNCcnt explicitly, consumers' DS
// loads are drained by register dependencies before they reach the barrier.
__device__ __forceinline__ void gcn_wg_barrier() {
#if GCN_ASYNC_LDS
  asm volatile("s_barrier_signal -1\n\ts_barrier_wait -1" ::: "memory");
#else
  __syncthreads();
#endif
}

// ---------------- COO SpMM scatter: Y[r] += v * X[c] ----------------
// One wave per edge: 32 lanes x float2 covers the 64-wide (256B) feature row.
// Edge lists are double-buffered through LDS via async global->LDS loads.
// (Placed first in the TU so the disasm snippet shows this kernel.)
__global__ void __launch_bounds__(GCN_SCATTER_BLOCK)
gcn_spmm_scatter(const int* __restrict__ rows, const int* __restrict__ cols,
                 const float* __restrict__ vals, const float* __restrict__ X,
                 float* __restrict__ Y, int nnz) {
  __shared__ int   rows_s[2][GCN_CHUNK];
  __shared__ int   cols_s[2][GCN_CHUNK];
  __shared__ float vals_s[2][GCN_CHUNK];

  const int tid  = threadIdx.x;
  const int lane = tid & 31;
  const int wave = tid >> 5;
  const int stride = (int)gridDim.x * GCN_CHUNK;
  int base = (int)blockIdx.x * GCN_CHUNK;
  if (base >= nnz) return;  // block-uniform exit, no barrier hazard

  // Preload chunk 0 into buffer 0 (addresses clamped -> full-EXEC issue,
  // so per-wave ASYNCcnt bookkeeping is exact: 3 ops per wave per chunk).
  {
    int idx = base + tid; if (idx >= nnz) idx = nnz - 1;
#if GCN_ASYNC_LDS
    gcn_async_b32(&rows[idx], &rows_s[0][tid]);
    gcn_async_b32(&cols[idx], &cols_s[0][tid]);
    gcn_async_b32(&vals[idx], &vals_s[0][tid]);
#else
    rows_s[0][tid] = rows[idx];
    cols_s[0][tid] = cols[idx];
    vals_s[0][tid] = vals[idx];
#endif
  }

  int cur = 0;
  for (; base < nnz; base += stride) {
    const int nbase = base + stride;
    const bool have_next = (nbase < nnz);  // block-uniform
    if (have_next) {
      int idx = nbase + tid; if (idx >= nnz) idx = nnz - 1;
#if GCN_ASYNC_LDS
      gcn_async_b32(&rows[idx], &rows_s[cur ^ 1][tid]);
      gcn_async_b32(&cols[idx], &cols_s[cur ^ 1][tid]);
      gcn_async_b32(&vals[idx], &vals_s[cur ^ 1][tid]);
#else
      rows_s[cur ^ 1][tid] = rows[idx];
      cols_s[cur ^ 1][tid] = cols[idx];
      vals_s[cur ^ 1][tid] = vals[idx];
#endif
    }
#if GCN_ASYNC_LDS
    // Async ops complete in order: waiting down to the 3 in-flight next-chunk
    // ops guarantees the current buffer is fully resident for this wave.
    if (have_next) __builtin_amdgcn_s_wait_asynccnt(3);
    else           __builtin_amdgcn_s_wait_asynccnt(0);
#endif
    gcn_wg_barrier();

    int count = nnz - base; if (count > GCN_CHUNK) count = GCN_CHUNK;
    const int*   rs = rows_s[cur];
    const int*   cs = cols_s[cur];
    const float* vs = vals_s[cur];

    if (count == GCN_CHUNK) {
      // Fast path (every chunk when nnz % CHUNK == 0): no per-edge bounds checks.
      #pragma unroll 4
      for (int j = 0; j < 32; ++j) {
        const int e = wave * 32 + j;
        const int   c = cs[e];   // LDS broadcast reads
        const int   r = rs[e];
        const float v = vs[e];

        // Prefetch a future gather row (lanes 0 and 16 cover the 256B row).
        if ((lane & 15) == 0) {
          int epf = e + 8; if (epf >= GCN_CHUNK) epf = GCN_CHUNK - 1;
          __builtin_prefetch(&X[(size_t)cs[epf] * GCN_EMB + (size_t)(lane << 1)], 0, 1);
        }

        const float2* xr = (const float2*)(X + (size_t)c * GCN_EMB);
        const float2  d  = xr[lane];                 // coalesced global_load_b64
        float* yr = Y + (size_t)r * GCN_EMB + (size_t)(lane << 1);
        atomicAdd(yr,     v * d.x);                  // global_atomic_add_f32 (L2)
        atomicAdd(yr + 1, v * d.y);
      }
    } else {
      for (int j = 0; j < 32; ++j) {
        const int e = wave * 32 + j;
        if (e >= count) break;
        const int   c = cs[e];
        const int   r = rs[e];
        const float v = vs[e];
        const float2* xr = (const float2*)(X + (size_t)c * GCN_EMB);
        const float2  d  = xr[lane];
        float* yr = Y + (size_t)r * GCN_EMB + (size_t)(lane << 1);
        atomicAdd(yr,     v * d.x);
        atomicAdd(yr + 1, v * d.y);
      }
    }
    gcn_wg_barrier();  // everyone done reading buf[cur] before it is reused
    cur ^= 1;
  }
}

// ---------------- init: X = e, out = e, Y = 0 (float4) ----------------
__global__ void __launch_bounds__(256)
gcn_init_kernel(const float4* __restrict__ user4, const float4* __restrict__ item4,
                float4* __restrict__ X4, float4* __restrict__ Y4,
                float4* __restrict__ out4) {
  const size_t n4 = (size_t)GCN_N_NODES * GCN_EMB / 4;
  const size_t u4 = (size_t)GCN_N_USERS * GCN_EMB / 4;
  size_t i = (size_t)blockIdx.x * blockDim.x + threadIdx.x;
  if (i >= n4) return;
  float4 e = (i < u4) ? user4[i] : item4[i - u4];
  X4[i]   = e;
  out4[i] = e;
  Y4[i]   = make_float4(0.f, 0.f, 0.f, 0.f);
}

// ---------------- acc += layer_out (and zero next destination) ----------------
__global__ void __launch_bounds__(256)
gcn_accum_kernel(float4* __restrict__ out4, const float4* __restrict__ src4,
                 float4* __restrict__ zero4, float scale, int do_zero) {
  const size_t n4 = (size_t)GCN_N_NODES * GCN_EMB / 4;
  size_t i = (size_t)blockIdx.x * blockDim.x + threadIdx.x;
  if (i >= n4) return;
  float4 a = out4[i];
  float4 y = src4[i];
  a.x = (a.x + y.x) * scale;
  a.y = (a.y + y.y) * scale;
  a.z = (a.z + y.z) * scale;
  a.w = (a.w + y.w) * scale;
  out4[i] = a;
  if (do_zero) zero4[i] = make_float4(0.f, 0.f, 0.f, 0.f);
}

extern "C" void kernel_launch(void* const* d_in, const int* in_sizes, int n_in,
                              void* d_out, int out_size, void* d_ws, size_t ws_size,
                              hipStream_t stream) {
  const float* user = (const float*)d_in[0];
  const float* item = (const float*)d_in[1];
  const int*   rows = (const int*)d_in[2];
  const int*   cols = (const int*)d_in[3];
  const float* vals = (const float*)d_in[4];
  const int    nnz  = in_sizes[2];

  float* out = (float*)d_out;
  float* X   = (float*)d_ws;                       // cur buffer A (128 MB)
  float* Y   = X + (size_t)GCN_N_NODES * GCN_EMB;  // cur buffer B (128 MB)

  const size_t n4 = (size_t)GCN_N_NODES * GCN_EMB / 4;
  const dim3 blk(256);
  const dim3 grid_e((unsigned)((n4 + 255) / 256));

  gcn_init_kernel<<<grid_e, blk, 0, stream>>>(
      (const float4*)user, (const float4*)item,
      (float4*)X, (float4*)Y, (float4*)out);

  int chunks = (nnz + GCN_CHUNK - 1) / GCN_CHUNK;
  int sgrid  = chunks < 4096 ? chunks : 4096;  // grid-stride -> double buffering engages

  // Layer 1: X -> Y ; acc += Y ; zero X
  gcn_spmm_scatter<<<sgrid, GCN_SCATTER_BLOCK, 0, stream>>>(rows, cols, vals, X, Y, nnz);
  gcn_accum_kernel<<<grid_e, blk, 0, stream>>>((float4*)out, (const float4*)Y,
                                               (float4*)X, 1.0f, 1);
  // Layer 2: Y -> X ; acc += X ; zero Y
  gcn_spmm_scatter<<<sgrid, GCN_SCATTER_BLOCK, 0, stream>>>(rows, cols, vals, Y, X, nnz);
  gcn_accum_kernel<<<grid_e, blk, 0, stream>>>((float4*)out, (const float4*)X,
                                               (float4*)Y, 1.0f, 1);
  // Layer 3: X -> Y ; acc = (acc + Y) * 0.25
  gcn_spmm_scatter<<<sgrid, GCN_SCATTER_BLOCK, 0, stream>>>(rows, cols, vals, X, Y, nnz);
  gcn_accum_kernel<<<grid_e, blk, 0, stream>>>((float4*)out, (const float4*)Y,
                                               (float4*)Y, 0.25f, 0);
}